// PhysDiffCamera_49555332662108
// MI455X (gfx1250) — compile-verified
//
#include <hip/hip_runtime.h>

#define HH 2048
#define WW 2048
#define NK 22             // k in [0, 21]
#define HM 10             // max_coc // 2
#define KW 21             // window width (2*HM+1)
#define TAPS (KW * KW)    // 441
#define GOFF 63           // guard band entries each side of a k-slice (3 rows * 21)
#define WSTR (TAPS + 2 * GOFF) // 567: guarded stride per k-slice
#define TS 32             // output tile edge
#define IT2 (TS + 2 * HM) // 52: input tile edge with halo
#define NTHREADS 256
#define JB 4              // outputs per thread (vertical register blocking)

typedef unsigned int u32;
typedef unsigned long long u64;
typedef __attribute__((ext_vector_type(4))) u32 u32x4_t;
typedef __attribute__((ext_vector_type(4))) int i32x4_t;
typedef __attribute__((ext_vector_type(8))) int i32x8_t;

// Issue one 2D TDM tile load (global -> LDS), 4-byte elements.
// tensor dims == tile dims: caller guarantees the tile is fully in-bounds,
// so the OOB logic never triggers regardless of interpretation.
__device__ __forceinline__ void tdm_load_2d_f32(u32 lds_byte, u64 gaddr,
                                                u32 tile_d0, u32 tile_d1,
                                                u64 stride0_elems) {
  u32x4_t g0;
  g0.x = 1u;                                   // count=1, is_restore=0, gather=0
  g0.y = lds_byte;                             // [63:32] lds_addr (bytes)
  g0.z = (u32)gaddr;                           // [95:64] global_addr[31:0]
  g0.w = (u32)((gaddr >> 32) & 0x01FFFFFFull)  // [120:96] global_addr[56:32]
         | (2u << 30);                         // [127:126] type=2 (image)

  i32x8_t g1;
  g1[0] = (int)(2u << 16);                     // workgroup_mask=0, data_size=2 (4B)
  g1[1] = (int)((tile_d0 & 0xFFFFu) << 16);    // atomic_barrier_addr=0 | tensor_dim0 lo16
  g1[2] = (int)(((tile_d0 >> 16) & 0xFFFFu) |
                ((tile_d1 & 0xFFFFu) << 16));  // tensor_dim0 hi16 | tensor_dim1 lo16
  g1[3] = (int)(((tile_d1 >> 16) & 0xFFFFu) |
                ((tile_d0 & 0xFFFFu) << 16));  // tensor_dim1 hi16 | tile_dim0
  g1[4] = (int)(tile_d1 & 0xFFFFu);            // tile_dim1 | tile_dim2=0
  g1[5] = (int)(u32)stride0_elems;             // tensor_dim0_stride[31:0]
  g1[6] = (int)(u32)((stride0_elems >> 32) & 0xFFFFull); // stride0 hi | stride1 lo=0
  g1[7] = 0;                                   // stride1 hi = 0

  i32x4_t z4 = {0, 0, 0, 0};
#if defined(__clang_major__) && (__clang_major__ >= 23)
  i32x8_t z8 = {0, 0, 0, 0, 0, 0, 0, 0};
  __builtin_amdgcn_tensor_load_to_lds(g0, g1, z4, z4, z8, 0);
#else
  __builtin_amdgcn_tensor_load_to_lds(g0, g1, z4, z4, 0);
#endif
}

__global__ __launch_bounds__(NTHREADS)
void PhysDiffCamera_defocus_kernel(const float* __restrict__ img,
                                   const int* __restrict__ ks,
                                   float* __restrict__ out) {
  __shared__ float wtab[NK * WSTR];        // 49896 B: guarded gaussian table
  __shared__ float tileRGB[IT2 * IT2 * 3]; // 32448 B
  __shared__ int   ktile[IT2 * IT2];       // 10816 B (raw k, then k*WSTR+GOFF)

  const int tid = threadIdx.x;
  const int bx = blockIdx.x, by = blockIdx.y;
  const int tx0 = bx * TS - HM;
  const int ty0 = by * TS - HM;
  const bool interior = (bx >= 1) && (bx <= (WW / TS) - 2) &&
                        (by >= 1) && (by <= (HH / TS) - 2);

  if (interior) {
    // One wave issues two TDM tile loads; DMA overlaps table computation below.
    if (tid < 32) {
      u64 gimg = (u64)(uintptr_t)(const void*)(img + ((u64)ty0 * WW + tx0) * 3);
      u64 gks  = (u64)(uintptr_t)(const void*)(ks  + ((u64)ty0 * WW + tx0));
      u32 lds_rgb = (u32)(uintptr_t)(void*)(&tileRGB[0]); // low 32b of flat = LDS offset
      u32 lds_k   = (u32)(uintptr_t)(void*)(&ktile[0]);
      tdm_load_2d_f32(lds_rgb, gimg, IT2 * 3, IT2, (u64)WW * 3);
      tdm_load_2d_f32(lds_k,   gks,  IT2,     IT2, (u64)WW);
    }
  } else {
    // Border tiles: cooperative clamped load; OOB sources get rgb=0 so they
    // contribute nothing regardless of weight.
    for (int i = tid; i < IT2 * IT2; i += NTHREADS) {
      int ly = i / IT2, lx = i - ly * IT2;
      int gy = ty0 + ly, gx = tx0 + lx;
      float r = 0.f, g = 0.f, b = 0.f;
      int kk = 0;
      if (gy >= 0 && gy < HH && gx >= 0 && gx < WW) {
        u64 gi = (u64)gy * WW + gx;
        r = img[gi * 3 + 0];
        g = img[gi * 3 + 1];
        b = img[gi * 3 + 2];
        kk = ks[gi];
      }
      tileRGB[i * 3 + 0] = r;
      tileRGB[i * 3 + 1] = g;
      tileRGB[i * 3 + 2] = b;
      ktile[i] = kk;
    }
  }

  // Guarded weight table: slice k occupies [k*WSTR, (k+1)*WSTR), live taps at
  // [GOFF, GOFF+441), zeros elsewhere. Overlaps with the TDM transfer.
  for (int i = tid; i < NK * WSTR; i += NTHREADS) {
    int k = i / WSTR;
    int t = i - k * WSTR - GOFF;
    float w = 0.0f;
    if (t >= 0 && t < TAPS) {
      int ty_ = t / KW;
      int dy = ty_ - HM;
      int dx = (t - ty_ * KW) - HM;
      if (k <= 1) {
        w = (dy == 0 && dx == 0) ? 1.0f : 0.0f;
      } else {
        int hk = k >> 1;
        if (dy >= -hk && dy <= hk && dx >= -hk && dx <= hk) {
          float s  = (float)k * (1.0f / 6.0f);
          float s2 = s * s;
          float inv2s2 = 0.5f / s2;
          float norm   = 1.0f / (6.283185307179586f * s2);
          float d2 = (float)(dy * dy + dx * dx);
          w = __expf(-d2 * inv2s2) * norm;
        }
      }
    }
    wtab[i] = w;
  }

  if (interior && tid < 32) {
    __builtin_amdgcn_s_wait_tensorcnt(0);  // wave 0 drains its TDM ops
  }
  __syncthreads();

  // In-place transform: raw k -> guarded table base index (clamped defensively).
  for (int i = tid; i < IT2 * IT2; i += NTHREADS) {
    int k = ktile[i];
    k = (k < 0) ? 0 : ((k > (NK - 1)) ? (NK - 1) : k);
    ktile[i] = k * WSTR + GOFF;
  }
  __syncthreads();

  // Gather with 4-way vertical register blocking.
  // Thread (tx, tyq) produces outputs (by*32 + tyq*4 + j, bx*32 + tx), j=0..3.
  // Source union window: tile rows [tyq*4, tyq*4+24), tile cols [tx, tx+21).
  // For source (r, c): dy_j = (j+10)-r, dx = 10-c, guarded tap index =
  //   GOFF + (j+20-r)*21 + (20-c)  in [0, 567) for all r in [0,24), j in [0,4).
  // The 12 independent scalar FMAs per source are ideal VOPD dual-issue fodder.
  const int tx  = tid & (TS - 1);
  const int tyq = tid >> 5;
  float acc[JB][3];
#pragma unroll
  for (int j = 0; j < JB; ++j) { acc[j][0] = 0.f; acc[j][1] = 0.f; acc[j][2] = 0.f; }

#pragma unroll 1
  for (int r = 0; r < KW + JB - 1; ++r) {        // 24 source rows
    const int pBase = (tyq * JB + r) * IT2 + tx; // first source pixel this row
    const int rowC  = (20 - r) * KW;             // may be negative; net index >= 0
#pragma unroll
    for (int c = 0; c < KW; ++c) {
      const int p  = pBase + c;
      const float rr = tileRGB[p * 3 + 0];
      const float gg = tileRGB[p * 3 + 1];
      const float bb = tileRGB[p * 3 + 2];
      const int wb = ktile[p] + rowC;            // guarded slice base + row offset
#pragma unroll
      for (int j = 0; j < JB; ++j) {
        const float w = wtab[wb + j * KW + (KW - 1 - c)];
        acc[j][0] = fmaf(rr, w, acc[j][0]);
        acc[j][1] = fmaf(gg, w, acc[j][1]);
        acc[j][2] = fmaf(bb, w, acc[j][2]);
      }
    }
  }

  const int gx  = bx * TS + tx;
  const int gy0 = by * TS + tyq * JB;
#pragma unroll
  for (int j = 0; j < JB; ++j) {
    u64 oi = ((u64)(gy0 + j) * WW + (u64)gx) * 3;
    out[oi + 0] = acc[j][0];
    out[oi + 1] = acc[j][1];
    out[oi + 2] = acc[j][2];
  }
}

extern "C" void kernel_launch(void* const* d_in, const int* in_sizes, int n_in,
                              void* d_out, int out_size, void* d_ws, size_t ws_size,
                              hipStream_t stream) {
  (void)in_sizes; (void)n_in; (void)out_size; (void)d_ws; (void)ws_size;
  const float* img = (const float*)d_in[0];
  const int*   ks  = (const int*)d_in[1];
  // d_in[2] = max_coc (= 21) is baked into the constants above.
  float* out = (float*)d_out;
  dim3 grid(WW / TS, HH / TS);
  PhysDiffCamera_defocus_kernel<<<grid, NTHREADS, 0, stream>>>(img, ks, out);
}